// dist_loss_70952859730215
// MI455X (gfx1250) — compile-verified
//
#include <hip/hip_runtime.h>
#include <math.h>

#define BATCH 8
#define CCH   32
#define KLAB  64
#define CHUNK 4096
#define DDIST 1.5f

typedef __attribute__((ext_vector_type(2))) float v2f;
typedef __attribute__((ext_vector_type(8))) float v8f;

// Workspace layout (floats):
//   [0, B*C*K)            per-batch segment sums, [b][c][k]
//   [B*C*K, +B*K)         per-batch label counts, [b][k]
//   [.., +B)              per-batch losses

__global__ void zero_kernel(float* __restrict__ ws, int total) {
  int i = blockIdx.x * blockDim.x + threadIdx.x;
  if (i < total) ws[i] = 0.0f;
}

// Phase 1: streaming segment-sum. Grid: (ceil(N/CHUNK), B), 256 threads.
__global__ __launch_bounds__(256) void segsum_kernel(
    const float* __restrict__ pred, const int* __restrict__ label,
    float* __restrict__ ws_sums, float* __restrict__ ws_cnt, int N) {
  __shared__ float s_sums[CCH * KLAB];  // [c][k]: bank index == label
  __shared__ float s_cnt[KLAB];
  const int tid = threadIdx.x;
  const int b   = blockIdx.y;
  const int n0  = blockIdx.x * CHUNK;

  for (int i = tid; i < CCH * KLAB; i += 256) s_sums[i] = 0.0f;
  if (tid < KLAB) s_cnt[tid] = 0.0f;
  __syncthreads();

  const int*   lab_b  = label + (size_t)b * N;
  const float* pred_b = pred  + (size_t)b * CCH * N;

  int nend = n0 + CHUNK; if (nend > N) nend = N;
  for (int n = n0 + tid * 4; n < nend; n += 256 * 4) {
    if (n + 3 < nend) {
      int4 lv = *reinterpret_cast<const int4*>(lab_b + n);
      int l0 = lv.x & 63, l1 = lv.y & 63, l2 = lv.z & 63, l3 = lv.w & 63;
      atomicAdd(&s_cnt[l0], 1.0f);
      atomicAdd(&s_cnt[l1], 1.0f);
      atomicAdd(&s_cnt[l2], 1.0f);
      atomicAdd(&s_cnt[l3], 1.0f);
#pragma unroll 4
      for (int c = 0; c < CCH; ++c) {
        float4 pv = *reinterpret_cast<const float4*>(pred_b + (size_t)c * N + n);
        float* row = s_sums + c * KLAB;
        atomicAdd(row + l0, pv.x);
        atomicAdd(row + l1, pv.y);
        atomicAdd(row + l2, pv.z);
        atomicAdd(row + l3, pv.w);
      }
    } else {
      for (int e = 0; e < 4; ++e) {
        int nn = n + e;
        if (nn < nend) {
          int l = lab_b[nn] & 63;
          atomicAdd(&s_cnt[l], 1.0f);
          for (int c = 0; c < CCH; ++c)
            atomicAdd(&s_sums[c * KLAB + l], pred_b[(size_t)c * N + nn]);
        }
      }
    }
  }
  __syncthreads();

  float* sums_g = ws_sums + (size_t)b * CCH * KLAB;
  for (int i = tid; i < CCH * KLAB; i += 256) atomicAdd(&sums_g[i], s_sums[i]);
  if (tid < KLAB) atomicAdd(&ws_cnt[(size_t)b * KLAB + tid], s_cnt[tid]);
}

// Phase 2: per-batch gram via V_WMMA_F32_16X16X4_F32 and hinge loss.
// Grid: (B), 128 threads = 4 waves, one 16x16 gram tile per wave.
__global__ __launch_bounds__(128) void gram_loss_kernel(
    const float* __restrict__ ws_sums, const float* __restrict__ ws_cnt,
    float* __restrict__ ws_loss) {
  __shared__ float s_center[KLAB * 33];  // [k][c], stride 33 (bank-conflict pad)
  __shared__ float s_gram[32 * 33];      // [i][j], stride 33
  __shared__ float s_cnt[KLAB];
  __shared__ float s_red[4];

  const int b   = blockIdx.x;
  const int tid = threadIdx.x;

  if (tid < KLAB) {
    float c = ws_cnt[(size_t)b * KLAB + tid];
    s_cnt[tid] = (c > 1.0f) ? c : 1.0f;  // max(counts, 1)
  }
  __syncthreads();

  for (int i = tid; i < CCH * KLAB; i += 128) {
    int c = i >> 6, k = i & 63;  // ws layout [c][k]
    s_center[k * 33 + c] = ws_sums[(size_t)b * CCH * KLAB + i] / s_cnt[k];
  }
  __syncthreads();

  const int wave = tid >> 5, lane = tid & 31;
  const int ci0 = (wave >> 1) * 16;  // tile row base (columns of center)
  const int cj0 = (wave & 1) * 16;   // tile col base
  const int lrow  = lane & 15;
  const int khalf = (lane >> 4) * 2;  // A/B: upper half-wave holds K=2,3

  v8f acc = {};
  for (int kb = 0; kb < KLAB; kb += 4) {
    int ka = kb + khalf;
    v2f av, bv;
    av.x = s_center[(ka + 0) * 33 + ci0 + lrow];
    av.y = s_center[(ka + 1) * 33 + ci0 + lrow];
    bv.x = s_center[(ka + 0) * 33 + cj0 + lrow];
    bv.y = s_center[(ka + 1) * 33 + cj0 + lrow];
    acc = __builtin_amdgcn_wmma_f32_16x16x4_f32(
        /*neg_a=*/false, av, /*neg_b=*/false, bv,
        /*c_mod=*/(short)0, acc, /*reuse_a=*/false, /*reuse_b=*/false);
  }

  // D layout: VGPR r -> M = r + 8*(lane>=16), N = lane%16
  const int m0 = (lane >> 4) * 8;
#pragma unroll
  for (int r = 0; r < 8; ++r)
    s_gram[(ci0 + m0 + r) * 33 + (cj0 + lrow)] = acc[r];
  __syncthreads();

  // hinge loss over full 32x32 (diag of gram == n2, matches reference incl. diagonal)
  float lsum = 0.0f;
  for (int e = tid; e < 32 * 32; e += 128) {
    int i = e >> 5, j = e & 31;
    float g  = s_gram[i * 33 + j];
    float sq = s_gram[i * 33 + i] + s_gram[j * 33 + j] - 2.0f * g;
    sq = sq > 0.0f ? sq : 0.0f;
    float dist = sq > 0.0f ? sqrtf(sq) : 0.0f;
    float h = 2.0f * DDIST - dist;
    h = h > 0.0f ? h : 0.0f;
    lsum += h * h;
  }
  for (int off = 16; off > 0; off >>= 1) lsum += __shfl_down(lsum, off, 32);
  if (lane == 0) s_red[wave] = lsum;
  __syncthreads();
  if (tid == 0) {
    float t = s_red[0] + s_red[1] + s_red[2] + s_red[3];
    ws_loss[b] = t / (2.0f * (float)KLAB * ((float)(KLAB - 1) + 1e-16f));
  }
}

// Phase 3: deterministic sum of per-batch losses.
__global__ void finalize_kernel(const float* __restrict__ ws_loss,
                                float* __restrict__ out) {
  if (threadIdx.x == 0 && blockIdx.x == 0) {
    float s = 0.0f;
    for (int b = 0; b < BATCH; ++b) s += ws_loss[b];
    out[0] = s;
  }
}

extern "C" void kernel_launch(void* const* d_in, const int* in_sizes, int n_in,
                              void* d_out, int out_size, void* d_ws, size_t ws_size,
                              hipStream_t stream) {
  const float* pred  = (const float*)d_in[0];
  const int*   label = (const int*)d_in[1];
  float* out = (float*)d_out;
  float* ws  = (float*)d_ws;

  const int B = BATCH;
  const int N = in_sizes[1] / B;  // label is [B, N]

  float* ws_sums = ws;
  float* ws_cnt  = ws + (size_t)B * CCH * KLAB;
  float* ws_loss = ws_cnt + (size_t)B * KLAB;

  int total = B * CCH * KLAB + B * KLAB + B;
  zero_kernel<<<(total + 255) / 256, 256, 0, stream>>>(ws, total);

  dim3 g1((N + CHUNK - 1) / CHUNK, B);
  segsum_kernel<<<g1, 256, 0, stream>>>(pred, label, ws_sums, ws_cnt, N);

  gram_loss_kernel<<<B, 128, 0, stream>>>(ws_sums, ws_cnt, ws_loss);

  finalize_kernel<<<1, 32, 0, stream>>>(ws_loss, out);
}